// Classifier1L_79697413144782
// MI455X (gfx1250) — compile-verified
//
#include <hip/hip_runtime.h>

typedef float v2f __attribute__((ext_vector_type(2)));
typedef float v8f __attribute__((ext_vector_type(8)));

#define BATCH  4096
#define EMBED  128
#define HIDDEN 256
#define NOUT   10
#define MAXLEN 200
#define XSTRIDE (MAXLEN + 1)

// ---------------------------------------------------------------------------
// Kernel 1: embedding-bag sum pooling. One wave32 per batch row.
// Each lane owns one float4 (4 consecutive floats) of the 128-wide embedding,
// so each token gather is one coalesced 512B/wave global_load_b128.
// Token id 0 contributes zero (reference sets table[0] = 0).
// ---------------------------------------------------------------------------
__global__ __launch_bounds__(256) void pool_kernel(const int* __restrict__ x,
                                                   const float* __restrict__ emb,
                                                   float* __restrict__ pooled) {
  const int row  = (int)((blockIdx.x * blockDim.x + threadIdx.x) >> 5);
  const int lane = (int)(threadIdx.x & 31);
  if (row >= BATCH) return;

  const int* __restrict__ xr = x + row * XSTRIDE;
  const int len = xr[0];
  const float4* __restrict__ embv = (const float4*)emb;  // 32 float4 per row

  float4 acc = {0.f, 0.f, 0.f, 0.f};
  int j = 0;
  // 4x unrolled: 4 independent gathers in flight per wave per iteration
  for (; j + 4 <= len; j += 4) {
    const int t0 = xr[1 + j], t1 = xr[2 + j], t2 = xr[3 + j], t3 = xr[4 + j];
    const float4 e0 = embv[t0 * 32 + lane];
    const float4 e1 = embv[t1 * 32 + lane];
    const float4 e2 = embv[t2 * 32 + lane];
    const float4 e3 = embv[t3 * 32 + lane];
    const float s0 = (t0 != 0) ? 1.f : 0.f;
    const float s1 = (t1 != 0) ? 1.f : 0.f;
    const float s2 = (t2 != 0) ? 1.f : 0.f;
    const float s3 = (t3 != 0) ? 1.f : 0.f;
    acc.x = fmaf(s0, e0.x, acc.x); acc.y = fmaf(s0, e0.y, acc.y);
    acc.z = fmaf(s0, e0.z, acc.z); acc.w = fmaf(s0, e0.w, acc.w);
    acc.x = fmaf(s1, e1.x, acc.x); acc.y = fmaf(s1, e1.y, acc.y);
    acc.z = fmaf(s1, e1.z, acc.z); acc.w = fmaf(s1, e1.w, acc.w);
    acc.x = fmaf(s2, e2.x, acc.x); acc.y = fmaf(s2, e2.y, acc.y);
    acc.z = fmaf(s2, e2.z, acc.z); acc.w = fmaf(s2, e2.w, acc.w);
    acc.x = fmaf(s3, e3.x, acc.x); acc.y = fmaf(s3, e3.y, acc.y);
    acc.z = fmaf(s3, e3.z, acc.z); acc.w = fmaf(s3, e3.w, acc.w);
  }
  for (; j < len; ++j) {
    const int t = xr[1 + j];
    const float4 e = embv[t * 32 + lane];
    const float s = (t != 0) ? 1.f : 0.f;
    acc.x = fmaf(s, e.x, acc.x); acc.y = fmaf(s, e.y, acc.y);
    acc.z = fmaf(s, e.z, acc.z); acc.w = fmaf(s, e.w, acc.w);
  }
  ((float4*)pooled)[row * 32 + lane] = acc;
}

// ---------------------------------------------------------------------------
// Kernel 2: fused MLP, fp32 WMMA (V_WMMA_F32_16X16X4_F32).
// One block (4 wave32) handles a 16-row tile of the batch.
//   Stage 1: H = relu(P @ W1 + b1), wave w computes cols [64w, 64w+64)
//   Stage 2: out = H @ W2 + b2, K=256 split across the 4 waves, LDS reduce
//
// WMMA f32 16x16x4 operand layout (wave32):
//   A: lane -> M = lane&15, elems {K = kh, kh+1}, kh = 2*(lane>>4)
//   B: lane -> N = lane&15, elems {K = kh, kh+1}
//   C/D: vgpr r, lane -> (M = r + 8*(lane>>4), N = lane&15)
// ---------------------------------------------------------------------------
#define PS_STRIDE 132   // 132 % 64 == 4  -> conflict-free column reads
#define HS_STRIDE 260   // 260 % 64 == 4  -> conflict-free col reads, row writes
#define PT_STRIDE 18    // 8*18 % 64 == 16 -> conflict-free half-wave writes

__global__ __launch_bounds__(128) void mlp_kernel(const float* __restrict__ pooled,
                                                  const float* __restrict__ W1,
                                                  const float* __restrict__ b1,
                                                  const float* __restrict__ W2,
                                                  const float* __restrict__ b2,
                                                  float* __restrict__ out) {
  __shared__ float Ps[16 * PS_STRIDE];          // pooled tile 16 x 128
  __shared__ float Hs[16 * HS_STRIDE];          // hidden tile 16 x 256
  __shared__ float Pt[4 * 16 * PT_STRIDE];      // per-wave 16x16 partials

  const int tid  = (int)threadIdx.x;
  const int lane = tid & 31;
  const int wave = tid >> 5;
  const int m0   = (int)blockIdx.x * 16;
  const int mn   = lane & 15;            // M for A-matrix, N for B/C
  const int kh   = (lane >> 4) << 1;     // 0 or 2: K-half within a step
  const int chi  = (lane >> 4) << 3;     // 0 or 8: C/D row offset

  // --- cooperative coalesced load of the 16x128 P tile into LDS ---
  for (int i = tid; i < 16 * 32; i += 128) {
    const int r = i >> 5, c = i & 31;
    const float4 v = ((const float4*)pooled)[(m0 + r) * 32 + c];
    float* dst = &Ps[r * PS_STRIDE + c * 4];
    dst[0] = v.x; dst[1] = v.y; dst[2] = v.z; dst[3] = v.w;
  }
  __syncthreads();

  // --- Stage 1: H[16 x 256] = relu(P @ W1 + b1); wave owns 4 N-tiles ---
  v8f acc[4] = {};
  const int nb = wave * 64;
  for (int k0 = 0; k0 < EMBED; k0 += 4) {
    v2f a;
    a.x = Ps[mn * PS_STRIDE + k0 + kh];
    a.y = Ps[mn * PS_STRIDE + k0 + kh + 1];
#pragma unroll
    for (int t = 0; t < 4; ++t) {
      v2f b;
      b.x = W1[(k0 + kh) * HIDDEN + nb + t * 16 + mn];
      b.y = W1[(k0 + kh + 1) * HIDDEN + nb + t * 16 + mn];
      acc[t] = __builtin_amdgcn_wmma_f32_16x16x4_f32(
          false, a, false, b, (short)0, acc[t], false, false);
    }
  }
#pragma unroll
  for (int t = 0; t < 4; ++t) {
#pragma unroll
    for (int r = 0; r < 8; ++r) {
      const int M = r + chi;
      const int N = nb + t * 16 + mn;
      const float h = acc[t][r] + b1[N];
      Hs[M * HS_STRIDE + N] = h > 0.f ? h : 0.f;
    }
  }
  __syncthreads();

  // --- Stage 2: out = H @ W2 + b2; K=256 split across the 4 waves ---
  v8f c2 = {};
  const int kb = wave * 64;
  for (int k0 = kb; k0 < kb + 64; k0 += 4) {
    v2f a;
    a.x = Hs[mn * HS_STRIDE + k0 + kh];
    a.y = Hs[mn * HS_STRIDE + k0 + kh + 1];
    v2f b;  // W2 zero-padded from N=10 to N=16 via lane predication
    b.x = (mn < NOUT) ? W2[(k0 + kh) * NOUT + mn] : 0.f;
    b.y = (mn < NOUT) ? W2[(k0 + kh + 1) * NOUT + mn] : 0.f;
    c2 = __builtin_amdgcn_wmma_f32_16x16x4_f32(
        false, a, false, b, (short)0, c2, false, false);
  }
#pragma unroll
  for (int r = 0; r < 8; ++r)
    Pt[wave * (16 * PT_STRIDE) + (r + chi) * PT_STRIDE + mn] = c2[r];
  __syncthreads();

  // --- reduce 4 partial tiles, add b2, write the 16x10 output tile ---
  for (int i = tid; i < 16 * NOUT; i += 128) {
    const int m = i / NOUT;
    const int n = i - m * NOUT;
    float s = b2[n];
#pragma unroll
    for (int w = 0; w < 4; ++w) s += Pt[w * (16 * PT_STRIDE) + m * PT_STRIDE + n];
    out[(m0 + m) * NOUT + n] = s;
  }
}

extern "C" void kernel_launch(void* const* d_in, const int* in_sizes, int n_in,
                              void* d_out, int out_size, void* d_ws, size_t ws_size,
                              hipStream_t stream) {
  const int*   x   = (const int*)d_in[0];
  const float* emb = (const float*)d_in[1];
  const float* W1  = (const float*)d_in[2];
  const float* b1  = (const float*)d_in[3];
  const float* W2  = (const float*)d_in[4];
  const float* b2  = (const float*)d_in[5];
  float* out    = (float*)d_out;
  float* pooled = (float*)d_ws;  // BATCH * EMBED floats = 2 MB scratch

  // 4096 rows, one wave32 per row, 8 waves per block
  pool_kernel<<<BATCH / 8, 256, 0, stream>>>(x, emb, pooled);
  // 16 rows per block
  mlp_kernel<<<BATCH / 16, 128, 0, stream>>>(pooled, W1, b1, W2, b2, out);
}